// TexRenderBatch_34789235097719
// MI455X (gfx1250) — compile-verified
//
#include <hip/hip_runtime.h>
#include <hip/hip_bf16.h>
#include <math.h>

#define BATCH   2
#define NPTS    600
#define NFACES  1024
#define IMH     80
#define IMW     80
#define TEXH    256
#define TEXW    256
#define SIGMA_F 7e-5f
#define DEGEPS  1e-8f

#define REC      32                 // floats per face record (128 B)
#define WCHUNK   32                 // faces per wave per round
#define NWAVES   4
#define RNDFACES (NWAVES * WCHUNK)  // 128 faces staged per round
#define ROUNDS   (NFACES / RNDFACES)        // 8
#define WFACES   (NFACES / NWAVES)          // 256 faces per wave

// out layout (floats): imrender[2,80,80,3] | improb[2,80,80] | normal1[2,1024,3] | mask[2,80,80]
#define OFF_IMPROB (BATCH * IMH * IMW * 3)            // 38400
#define OFF_NORMAL (OFF_IMPROB + BATCH * IMH * IMW)   // 51200
#define OFF_MASK   (OFF_NORMAL + BATCH * NFACES * 3)  // 57344

typedef float v2f __attribute__((ext_vector_type(2)));
typedef float v8f __attribute__((ext_vector_type(8)));
typedef int   v4i __attribute__((vector_size(16)));
typedef __attribute__((address_space(1))) v4i glob_v4i;
typedef __attribute__((address_space(3))) v4i lds_v4i;

#if __has_builtin(__builtin_amdgcn_wmma_f32_16x16x4_f32)
#define HAVE_WMMA_F32 1
#else
#define HAVE_WMMA_F32 0
#warning "missing __builtin_amdgcn_wmma_f32_16x16x4_f32 -> scalar fallback"
#endif

#if __has_builtin(__builtin_amdgcn_global_load_async_to_lds_b128)
#define HAVE_ASYNC 1
#else
#define HAVE_ASYNC 0
#warning "missing __builtin_amdgcn_global_load_async_to_lds_b128 -> sync LDS fill"
#endif

#if HAVE_ASYNC
#if __has_builtin(__builtin_amdgcn_s_wait_asynccnt)
#define WAIT_ASYNC(n) __builtin_amdgcn_s_wait_asynccnt(n)
#else
#define WAIT_ASYNC(n) asm volatile("s_wait_asynccnt %0" ::"i"(n) : "memory")
#endif
#else
#define WAIT_ASYNC(n)
#endif

// ---------------------------------------------------------------------------
// Face record (REC=32 floats, 128B):
//  [0..2]=w0 a,b,c [3]=0 | [4..6]=w1 a,b,c [7]=0 | [8..10]=z a,b,c [11]=0
//  [12..17]=x0,y0,x1,y1,x2,y2 | [18..19]=0
//  [20]=cx [21]=cy [22]=rad2 (negative => backfacing/degenerate) [23]=0
//  [24..29]=u0,v0,u1,v1,u2,v2 | [30..31]=0
// ---------------------------------------------------------------------------
__global__ void face_setup_kernel(const float* __restrict__ points,
                                  const int* __restrict__ faces,
                                  const float* __restrict__ camrot,
                                  const float* __restrict__ campos,
                                  const float* __restrict__ camproj,
                                  const float* __restrict__ uvp,
                                  float* __restrict__ ws,
                                  float* __restrict__ outNormal) {
  int gid = blockIdx.x * blockDim.x + threadIdx.x;
  if (gid >= BATCH * NFACES) return;
  int b = gid / NFACES;
  int f = gid - b * NFACES;

  const float* R = camrot + b * 9;
  const float* T = campos + b * 3;
  float prx = camproj[0], pry = camproj[1], prz = camproj[2];

  float cx[3], cy[3], cz[3], sx[3], sy[3], uu[3], vv[3];
#pragma unroll
  for (int k = 0; k < 3; ++k) {
    int pi = faces[f * 3 + k];
    const float* p = points + ((size_t)b * NPTS + pi) * 3;
    float dx = p[0] - T[0], dy = p[1] - T[1], dz = p[2] - T[2];
    float px_ = R[0] * dx + R[1] * dy + R[2] * dz;
    float py_ = R[3] * dx + R[4] * dy + R[5] * dz;
    float pz_ = R[6] * dx + R[7] * dy + R[8] * dz;
    cx[k] = px_; cy[k] = py_; cz[k] = pz_;
    float Xz = pz_ * prz;
    sx[k] = (px_ * prx) / Xz;
    sy[k] = (py_ * pry) / Xz;
    uu[k] = uvp[((size_t)b * NPTS + pi) * 2 + 0];
    vv[k] = uvp[((size_t)b * NPTS + pi) * 2 + 1];
  }

  // normal = cross(v1-v0, v2-v0); normal1 output
  float e1x = cx[1] - cx[0], e1y = cy[1] - cy[0], e1z = cz[1] - cz[0];
  float e2x = cx[2] - cx[0], e2y = cy[2] - cy[0], e2z = cz[2] - cz[0];
  float nx = e1y * e2z - e1z * e2y;
  float ny = e1z * e2x - e1x * e2z;
  float nz = e1x * e2y - e1y * e2x;
  float nrm = sqrtf(nx * nx + ny * ny + nz * nz) + 1e-10f;
  float* on = outNormal + (size_t)gid * 3;
  on[0] = nx / nrm; on[1] = ny / nrm; on[2] = nz / nrm;

  float denom = (sy[1] - sy[2]) * (sx[0] - sx[2]) + (sx[2] - sx[1]) * (sy[0] - sy[2]);
  bool deg = fabsf(denom) < DEGEPS;
  float inv = deg ? 0.f : (1.f / denom);

  float w0a = (sy[1] - sy[2]) * inv;
  float w0b = (sx[2] - sx[1]) * inv;
  float w0c = -(w0a * sx[2] + w0b * sy[2]);
  float w1a = (sy[2] - sy[0]) * inv;
  float w1b = (sx[0] - sx[2]) * inv;
  float w1c = -(w1a * sx[2] + w1b * sy[2]);
  float dz0 = cz[0] - cz[2], dz1 = cz[1] - cz[2];
  float za = w0a * dz0 + w1a * dz1;
  float zb = w0b * dz0 + w1b * dz1;
  float zc = cz[2] + w0c * dz0 + w1c * dz1;
  bool front = (nz > 0.f) && !deg;

  // bounding circle, expanded by the soft-coverage reach sqrt(80*sigma)~0.075
  float ccx = (sx[0] + sx[1] + sx[2]) * (1.f / 3.f);
  float ccy = (sy[0] + sy[1] + sy[2]) * (1.f / 3.f);
  float d0 = (sx[0] - ccx) * (sx[0] - ccx) + (sy[0] - ccy) * (sy[0] - ccy);
  float d1 = (sx[1] - ccx) * (sx[1] - ccx) + (sy[1] - ccy) * (sy[1] - ccy);
  float d2m = (sx[2] - ccx) * (sx[2] - ccx) + (sy[2] - ccy) * (sy[2] - ccy);
  float rad = sqrtf(fmaxf(d0, fmaxf(d1, d2m))) + 0.08f;
  float rad2 = front ? rad * rad : -1.f;

  float* r = ws + (size_t)gid * REC;
  r[0] = w0a; r[1] = w0b; r[2] = w0c; r[3] = 0.f;
  r[4] = w1a; r[5] = w1b; r[6] = w1c; r[7] = 0.f;
  r[8] = za;  r[9] = zb;  r[10] = zc; r[11] = 0.f;
  r[12] = sx[0]; r[13] = sy[0]; r[14] = sx[1]; r[15] = sy[1]; r[16] = sx[2]; r[17] = sy[2];
  r[18] = 0.f; r[19] = 0.f;
  r[20] = ccx; r[21] = ccy; r[22] = rad2; r[23] = 0.f;
  r[24] = uu[0]; r[25] = vv[0]; r[26] = uu[1]; r[27] = vv[1]; r[28] = uu[2]; r[29] = vv[2];
  r[30] = 0.f; r[31] = 0.f;
}

// ---------------------------------------------------------------------------
// Kernel 2: rasterize. 128 threads = 4 waves. All 4 waves share the SAME 16
// pixels; each wave owns a 256-face partition (4x latency hiding vs pixel
// split). Face records double-buffered in LDS via async global->LDS b128.
// w0/w1/z evaluated per 16px x 16face tile with V_WMMA_F32_16X16X4_F32.
// Partial (zmax/argmax, coverage, log-prob) merged via shfl + LDS.
// ---------------------------------------------------------------------------
__device__ __forceinline__ float edge_d2(float px, float py, float ax, float ay,
                                         float bx, float by) {
  float ex = bx - ax, ey = by - ay;
  float L2 = ex * ex + ey * ey + 1e-12f;
  float t = ((px - ax) * ex + (py - ay) * ey) * __builtin_amdgcn_rcpf(L2);
  t = fminf(fmaxf(t, 0.f), 1.f);
  float dx = px - (ax + t * ex);
  float dy = py - (ay + t * ey);
  return dx * dx + dy * dy;
}

// stage round j: faces {w*256 + j*32 + [0,32)} for w=0..3 -> lbuf[(w*32+local)*REC]
__device__ __forceinline__ void stage_round(const float* __restrict__ fbase,
                                            float* lbuf, int j, int tid) {
  // RNDFACES*REC*4 = 16384 B = 1024 x 16B segments, 128 threads -> 8 each
#pragma unroll
  for (int i = 0; i < 8; ++i) {
    int s = tid + i * 128;        // 0..1023
    int fIdx = s >> 3;            // buffer face slot 0..127
    int q = s & 7;                // 16B eighth (0..7) of the 128B record
    int w = fIdx >> 5;
    int local = fIdx & 31;
    const float* gs = fbase + ((size_t)(w * WFACES + j * WCHUNK + local)) * REC + q * 4;
    float* ls = lbuf + fIdx * REC + q * 4;
#if HAVE_ASYNC
    __builtin_amdgcn_global_load_async_to_lds_b128((glob_v4i*)gs, (lds_v4i*)ls, 0, 0);
#else
    float4 v = *(const float4*)gs;
    *(float4*)ls = v;
#endif
  }
}

__global__ __launch_bounds__(128) void raster_kernel(const float* __restrict__ ws,
                                                     const float* __restrict__ tex,
                                                     float* __restrict__ out) {
  __shared__ float lds[2][RNDFACES * REC];   // 2 x 16 KB
  __shared__ float mg[NWAVES][16][6];

  const int tid = threadIdx.x;
  const int lane = tid & 31;
  const int wave = tid >> 5;
  const int l15 = lane & 15;
  const bool hi = lane >= 16;

  const int blocksPerBatch = (IMH * IMW) / 16;  // 400
  const int b = blockIdx.x / blocksPerBatch;
  const int pix = (blockIdx.x % blocksPerBatch) * 16 + l15;
  const int row = pix / IMW;
  const int col = pix - row * IMW;
  const float px = ((float)col + 0.5f) / (float)IMW * 2.f - 1.f;
  const float py = 1.f - ((float)row + 0.5f) / (float)IMH * 2.f;

  const float* fbase = ws + (size_t)b * NFACES * REC;
  const int wbase = wave * WFACES;

  // B matrix (4x16): lanes 0-15 K=0,1 = (px,py); lanes 16-31 K=2,3 = (1,0)
  v2f bm;
  bm.x = hi ? 1.f : px;
  bm.y = hi ? 0.f : py;

  float zmax = -1e30f, bw0 = 0.f, bw1 = 0.f;
  int bface = 0, cov_any = 0;
  float logacc = 0.f;

  stage_round(fbase, &lds[0][0], 0, tid);

  for (int j = 0; j < ROUNDS; ++j) {
    if (j + 1 < ROUNDS) stage_round(fbase, &lds[(j + 1) & 1][0], j + 1, tid);
#if HAVE_ASYNC
    if (j + 1 < ROUNDS) { WAIT_ASYNC(8); } else { WAIT_ASYNC(0); }
#endif
    __syncthreads();
    const float* L = &lds[j & 1][0] + wave * WCHUNK * REC;

#pragma unroll
    for (int sub = 0; sub < WCHUNK / 16; ++sub) {
      // A (16x4): branch-free lo/hi loads: lanes 0-15 get (a,b), 16-31 get (c,0)
      const float* ra = L + (sub * 16 + l15) * REC + (hi ? 2 : 0);
      v2f aw0 = *(const v2f*)(ra + 0);
      v2f aw1 = *(const v2f*)(ra + 4);
      v2f az  = *(const v2f*)(ra + 8);
      v8f dw0 = {}, dw1 = {}, dz = {};
#if HAVE_WMMA_F32
      dw0 = __builtin_amdgcn_wmma_f32_16x16x4_f32(false, aw0, false, bm, (short)0, dw0, false, false);
      dw1 = __builtin_amdgcn_wmma_f32_16x16x4_f32(false, aw1, false, bm, (short)0, dw1, false, false);
      dz  = __builtin_amdgcn_wmma_f32_16x16x4_f32(false, az,  false, bm, (short)0, dz,  false, false);
#else
#pragma unroll
      for (int r = 0; r < 8; ++r) {
        const float* fr = L + (sub * 16 + r + (hi ? 8 : 0)) * REC;
        dw0[r] = fr[0] * px + fr[1] * py + fr[2];
        dw1[r] = fr[4] * px + fr[5] * py + fr[6];
        dz[r]  = fr[8] * px + fr[9] * py + fr[10];
      }
#endif
      // D layout: lane = my pixel; VGPR r = face (sub*16 + r) lo-lanes, (+8) hi-lanes
#pragma unroll
      for (int r = 0; r < 8; ++r) {
        const int fl = sub * 16 + r + (hi ? 8 : 0);
        const float* fr = L + fl * REC;
        float4 cc = *(const float4*)(fr + 20);  // cx, cy, rad2, 0
        float ddx = px - cc.x, ddy = py - cc.y;
        if (ddx * ddx + ddy * ddy <= cc.z) {    // inside reach & front-facing
          float w0 = dw0[r], w1 = dw1[r];
          float w2 = 1.f - w0 - w1;
          if ((w0 >= 0.f) && (w1 >= 0.f) && (w2 >= 0.f)) {
            cov_any = 1;
            float zv = dz[r];
            if (zv > zmax) { zmax = zv; bw0 = w0; bw1 = w1; bface = wbase + j * WCHUNK + fl; }
          } else {
            float x0 = fr[12], y0 = fr[13], x1 = fr[14], y1 = fr[15], x2 = fr[16], y2 = fr[17];
            float d2 = edge_d2(px, py, x0, y0, x1, y1);
            d2 = fminf(d2, edge_d2(px, py, x1, y1, x2, y2));
            d2 = fminf(d2, edge_d2(px, py, x2, y2, x0, y0));
            float t = -d2 * (1.f / SIGMA_F);
            if (t > -80.f) {  // exp underflows to 0 beyond this -> log1p(0)=0
              float prob = fminf(__expf(t), 1.f - 1e-7f);
              logacc += log1pf(-prob);
            }
          }
        }
      }
    }
    __syncthreads();
  }

  // merge hi/lo half-wave partitions (same pixel lives in lanes L and L+16)
  float zo  = __shfl_xor(zmax, 16, 32);
  float w0o = __shfl_xor(bw0, 16, 32);
  float w1o = __shfl_xor(bw1, 16, 32);
  int   fo  = __shfl_xor(bface, 16, 32);
  int   co  = __shfl_xor(cov_any, 16, 32);
  float ao  = __shfl_xor(logacc, 16, 32);
  if (zo > zmax) { zmax = zo; bw0 = w0o; bw1 = w1o; bface = fo; }
  cov_any |= co;
  logacc += ao;

  // merge the 4 face partitions across waves via LDS
  if (!hi) {
    mg[wave][l15][0] = zmax;
    mg[wave][l15][1] = bw0;
    mg[wave][l15][2] = bw1;
    mg[wave][l15][3] = __int_as_float(bface);
    mg[wave][l15][4] = (float)cov_any;
    mg[wave][l15][5] = logacc;
  }
  __syncthreads();

  if (tid < 16) {
    zmax = -1e30f; bw0 = 0.f; bw1 = 0.f; bface = 0; cov_any = 0; logacc = 0.f;
#pragma unroll
    for (int w = 0; w < NWAVES; ++w) {  // ascending face ranges: first-max wins ties
      float z = mg[w][tid][0];
      if (z > zmax) {
        zmax = z; bw0 = mg[w][tid][1]; bw1 = mg[w][tid][2];
        bface = __float_as_int(mg[w][tid][3]);
      }
      cov_any |= (mg[w][tid][4] != 0.f);
      logacc += mg[w][tid][5];
    }

    float improb = cov_any ? 1.f : (1.f - __expf(logacc));
    float maskv = cov_any ? 1.f : 0.f;
    float c0 = 0.f, c1 = 0.f, c2 = 0.f;
    if (cov_any) {
      const float* fu = fbase + (size_t)bface * REC + 24;
      float w2 = 1.f - bw0 - bw1;
      float u = bw0 * fu[0] + bw1 * fu[2] + w2 * fu[4];
      float v = bw0 * fu[1] + bw1 * fu[3] + w2 * fu[5];
      int tc = (int)fminf(fmaxf(rintf(u * (float)TEXW - 0.5f), 0.f), (float)(TEXW - 1));
      int tr = (int)fminf(fmaxf(rintf(v * (float)TEXH - 0.5f), 0.f), (float)(TEXH - 1));
      const float* tp = tex + (size_t)b * 3 * TEXH * TEXW + (size_t)tr * TEXW + tc;
      c0 = tp[0];
      c1 = tp[TEXH * TEXW];
      c2 = tp[2 * TEXH * TEXW];
    }
    const int po = b * IMH * IMW + pix;
    out[(size_t)po * 3 + 0] = c0;
    out[(size_t)po * 3 + 1] = c1;
    out[(size_t)po * 3 + 2] = c2;
    out[OFF_IMPROB + po] = improb;
    out[OFF_MASK + po] = maskv;
  }
}

extern "C" void kernel_launch(void* const* d_in, const int* in_sizes, int n_in,
                              void* d_out, int out_size, void* d_ws, size_t ws_size,
                              hipStream_t stream) {
  const float* points  = (const float*)d_in[0];
  const int*   faces   = (const int*)d_in[1];
  const float* camrot  = (const float*)d_in[2];
  const float* campos  = (const float*)d_in[3];
  const float* camproj = (const float*)d_in[4];
  const float* uv      = (const float*)d_in[5];
  const float* texture = (const float*)d_in[6];
  float* out = (float*)d_out;
  float* ws  = (float*)d_ws;  // BATCH*NFACES*REC*4 = 262144 bytes

  face_setup_kernel<<<(BATCH * NFACES + 255) / 256, 256, 0, stream>>>(
      points, faces, camrot, campos, camproj, uv, ws, out + OFF_NORMAL);

  raster_kernel<<<BATCH * ((IMH * IMW) / 16), 128, 0, stream>>>(ws, texture, out);
}